// NormalizedCrossEntropy_64029372449287
// MI455X (gfx1250) — compile-verified
//
#include <hip/hip_runtime.h>

typedef float f32x4 __attribute__((ext_vector_type(4)));
typedef float f32x2 __attribute__((ext_vector_type(2)));
typedef float f32x8 __attribute__((ext_vector_type(8)));

__device__ __forceinline__ float exp2_fast(float x) { return __builtin_amdgcn_exp2f(x); }
__device__ __forceinline__ float log2_fast(float x) { return __builtin_amdgcn_logf(x); }

#define TPB 256
#define NEG_BIG (-3.402823466e38f)

// One block (8 wave32s) per row. Single pass: online max/exp-sum in exp2 domain
// + plain row sum, 128-bit non-temporal streaming loads (dataset > L2).
__global__ __launch_bounds__(TPB) void nce_row_kernel(
    const float* __restrict__ logits,
    const int*   __restrict__ targets,
    float*       __restrict__ row_ratio,
    int C)
{
    const int row = blockIdx.x;
    const int tid = threadIdx.x;
    const size_t base = (size_t)row * (size_t)C;

    const float LOG2E = 1.4426950408889634f;
    const float LN2   = 0.6931471805599453f;

    float m = NEG_BIG;   // running max of y = x*log2e
    float s = 0.0f;      // running sum of 2^(y - m)
    float r = 0.0f;      // running plain sum of x

    const int C4 = C >> 2;
    const f32x4* p = reinterpret_cast<const f32x4*>(logits + base);

    for (int k = tid; k < C4; k += TPB) {
        f32x4 v = __builtin_nontemporal_load(&p[k]);   // global_load_b128, TH=NT
        r += (v.x + v.y) + (v.z + v.w);
        float y0 = v.x * LOG2E;
        float y1 = v.y * LOG2E;
        float y2 = v.z * LOG2E;
        float y3 = v.w * LOG2E;
        float mm = fmaxf(fmaxf(y0, y1), fmaxf(y2, y3));
        float nm = fmaxf(m, mm);
        s = s * exp2_fast(m - nm);                      // rescale old sum (==1 if max unchanged)
        m = nm;
        s += (exp2_fast(y0 - m) + exp2_fast(y1 - m)) +
             (exp2_fast(y2 - m) + exp2_fast(y3 - m));
    }
    // scalar tail (C % 4 != 0 case; C=10000 has none, kept for generality)
    for (int k = (C4 << 2) + tid; k < C; k += TPB) {
        float v = logits[base + k];
        r += v;
        float y  = v * LOG2E;
        float nm = fmaxf(m, y);
        s = s * exp2_fast(m - nm);
        m = nm;
        s += exp2_fast(y - m);
    }

    __shared__ float sm[TPB];
    __shared__ float ss[TPB];
    __shared__ float sr[TPB];
    sm[tid] = m; ss[tid] = s; sr[tid] = r;
    __syncthreads();

    for (int off = TPB / 2; off > 0; off >>= 1) {
        if (tid < off) {
            float m1 = sm[tid],        s1 = ss[tid];
            float m2 = sm[tid + off],  s2 = ss[tid + off];
            float nm = fmaxf(m1, m2);
            ss[tid] = s1 * exp2_fast(m1 - nm) + s2 * exp2_fast(m2 - nm);
            sm[tid] = nm;
            sr[tid] += sr[tid + off];
        }
        __syncthreads();
    }

    if (tid == 0) {
        float lse    = (sm[0] + log2_fast(ss[0])) * LN2;        // natural-log LSE
        float tgt    = logits[base + (size_t)targets[row]];
        float ce     = lse - tgt;
        float sum_ce = (float)C * lse - sr[0];
        row_ratio[row] = ce / (sum_ce + 1e-8f);
    }
}

// 8192 -> 1 mean. LDS tree to 64 partials, then V_WMMA_F32_16X16X4_F32 with
// B = ones(4x16) collapses the 64 values (as the 16x4 A matrix) to 16 row-sums
// in one matrix op; one shfl_xor finishes the reduction.
__global__ __launch_bounds__(TPB) void mean_wmma_kernel(
    const float* __restrict__ vals,
    float*       __restrict__ out,
    int N)
{
    const int tid = threadIdx.x;
    float acc = 0.0f;
    for (int i = tid; i < N; i += TPB) acc += vals[i];

    __shared__ float lds[TPB];
    lds[tid] = acc;
    __syncthreads();
    if (tid < 128) lds[tid] += lds[tid + 128];
    __syncthreads();
    if (tid < 64)  lds[tid] += lds[tid + 64];
    __syncthreads();

    if (tid < 32) {   // whole wave 0 enters -> EXEC all ones (WMMA requirement)
        // A (16x4 f32, 2 VGPRs): lane L<16 holds A[L][0..1], lane L>=16 holds A[L-16][2..3].
        // Mapping lds[L], lds[L+32] -> A is a bijection over all 64 partials.
        f32x2 a; a.x = lds[tid]; a.y = lds[tid + 32];
        f32x2 b; b.x = 1.0f;     b.y = 1.0f;          // B = ones(4x16)
        f32x8 c = {};
        c = __builtin_amdgcn_wmma_f32_16x16x4_f32(
                /*neg_a=*/false, a, /*neg_b=*/false, b,
                /*c_mod=*/(short)0, c, /*reuse_a=*/false, /*reuse_b=*/false);
        // D vgpr j: lanes 0-15 -> M=j, lanes 16-31 -> M=8+j. Sum the 8 accs,
        // then fold lanes 0..15 (rows 0-7) with lanes 16..31 (rows 8-15).
        float part = ((c[0] + c[1]) + (c[2] + c[3])) + ((c[4] + c[5]) + (c[6] + c[7]));
        part += __shfl_xor(part, 16, 32);
        if (tid == 0) out[0] = part / (float)N;
    }
}

extern "C" void kernel_launch(void* const* d_in, const int* in_sizes, int n_in,
                              void* d_out, int out_size, void* d_ws, size_t ws_size,
                              hipStream_t stream)
{
    const float* logits  = (const float*)d_in[0];
    const int*   targets = (const int*)d_in[1];

    const int B = in_sizes[1];              // 8192 rows
    const int C = in_sizes[0] / B;          // 10000 classes

    float* row_ratio = (float*)d_ws;        // B floats of scratch
    float* out       = (float*)d_out;       // single f32 result

    nce_row_kernel<<<B, TPB, 0, stream>>>(logits, targets, row_ratio, C);
    mean_wmma_kernel<<<1, TPB, 0, stream>>>(row_ratio, out, B);
}